// Texture_trans_16595753631954
// MI455X (gfx1250) — compile-verified
//
#include <hip/hip_runtime.h>
#include <math.h>

#define B_   4
#define C_   64
#define L_   4096          // 64*64
#define K9   576           // C*9
#define PAIRS 288          // K9/2
#define LT_  256           // L/16
#define BCL  (B_*C_*L_)    // 1,048,576 floats
#define BL   (B_*L_)       // 16,384
#define SEG  4
#define MT   4             // concurrent row tiles per wave
#define NT   2             // concurrent column tiles per wave

typedef __attribute__((ext_vector_type(2))) float v2f;
typedef __attribute__((ext_vector_type(8))) float v8f;

// ---------------- [B,L,C] -> [B,C,L] transpose via padded LDS tile ----------------
__global__ void k_transpose(const float* __restrict__ in, float* __restrict__ out) {
    __shared__ float tile[64 * 65];
    int b = blockIdx.y;
    int l0 = blockIdx.x * 64;
    int t = threadIdx.x;
    #pragma unroll
    for (int rep = 0; rep < 16; ++rep) {
        int e = rep * 256 + t;
        int i = e >> 6, c = e & 63;                       // coalesced read over c
        tile[i * 65 + c] = in[((size_t)b * L_ + l0 + i) * C_ + c];
    }
    __syncthreads();
    #pragma unroll
    for (int rep = 0; rep < 16; ++rep) {
        int e = rep * 256 + t;
        int c = e >> 6, i = e & 63;                       // coalesced write over l
        out[((size_t)b * C_ + c) * L_ + l0 + i] = tile[i * 65 + c];
    }
}

__global__ void k_sub(const float* __restrict__ a, const float* __restrict__ bb,
                      float* __restrict__ o, int n) {
    int i = blockIdx.x * 256 + threadIdx.x;
    if (i < n) o[i] = a[i] - bb[i];
}

// ---------------- 3x3-patch L2 norm -> reciprocal, per position ----------------
__global__ void k_patchnorm(const float* __restrict__ chw, float* __restrict__ inv) {
    int i = blockIdx.x * 256 + threadIdx.x;               // over B*L
    int b = i >> 12, l = i & (L_ - 1);
    int y = l >> 6, x = l & 63;
    float ss = 0.f;
    for (int c = 0; c < C_; ++c) {
        const float* img = chw + ((size_t)b * C_ + c) * L_;
        #pragma unroll
        for (int ky = 0; ky < 3; ++ky) {
            int iy = y + ky - 1; if ((unsigned)iy >= 64u) continue;
            #pragma unroll
            for (int kx = 0; kx < 3; ++kx) {
                int ix = x + kx - 1; if ((unsigned)ix >= 64u) continue;
                float v = img[iy * 64 + ix];
                ss += v * v;
            }
        }
    }
    inv[i] = 1.0f / fmaxf(sqrtf(ss), 1e-12f);
}

// ------- pack unfold into WMMA-lane order: U[b][lt][pair][16][2], optional row scale -------
__global__ void k_pack(const float* __restrict__ chw, const float* __restrict__ invOrNull,
                       float* __restrict__ U) {
    int b = blockIdx.y, lt = blockIdx.x;
    int t = threadIdx.x;
    int M = t & 15, pg = t >> 4;                          // 16 pair-groups of 18
    int l = lt * 16 + M;
    int y = l >> 6, x = l & 63;
    float sc = invOrNull ? invOrNull[b * L_ + l] : 1.0f;
    float2* out = (float2*)U + (size_t)(b * LT_ + lt) * PAIRS * 16;
    for (int pp = 0; pp < 18; ++pp) {
        int p = pg * 18 + pp;
        float v[2];
        #pragma unroll
        for (int j = 0; j < 2; ++j) {
            int k = 2 * p + j;
            int c = k / 9, t9 = k % 9;
            int ki = t9 / 3, kj = t9 % 3;
            int iy = y + ki - 1, ix = x + kj - 1;
            float vv = 0.f;
            if ((unsigned)iy < 64u && (unsigned)ix < 64u)
                vv = chw[((size_t)b * C_ + c) * L_ + iy * 64 + ix];
            v[j] = vv * sc;
        }
        out[p * 16 + M] = make_float2(v[0], v[1]);        // coalesced b64 store
    }
}

// ---------------- fused correlation GEMM (fp32 WMMA) + running max/argmax ----------------
// Each wave: 2 column tiles (32 cols) x 4 concurrent row tiles -> 8 WMMA chains,
// 6 b64 loads per 8 WMMAs (A operands shared across both column tiles).
__global__ void __launch_bounds__(128)
k_corr(const float* __restrict__ U0, const float* __restrict__ U1,
       float* __restrict__ pBest, int* __restrict__ pIdx) {
    int wave = threadIdx.x >> 5;
    int lane = threadIdx.x & 31;
    int half = lane >> 4;
    int n16  = lane & 15;
    int b    = blockIdx.z;
    int seg  = blockIdx.y;
    int mT0  = (blockIdx.x * 4 + wave) * NT;              // first of 2 column tiles

    const v2f* Bp0 = (const v2f*)U1 + (size_t)(b * LT_ + mT0) * PAIRS * 16 + n16;
    const v2f* Bp1 = Bp0 + (size_t)PAIRS * 16;
    float best0 = -3.0e38f, best1 = -3.0e38f;
    int bi0 = 0, bi1 = 0;

    for (int lt0 = seg * (LT_ / SEG); lt0 < (seg + 1) * (LT_ / SEG); lt0 += MT) {
        const v2f* A0 = (const v2f*)U0 + (size_t)(b * LT_ + lt0) * PAIRS * 16 + n16;
        v8f acc[NT][MT];
        #pragma unroll
        for (int n = 0; n < NT; ++n)
            #pragma unroll
            for (int t = 0; t < MT; ++t) acc[n][t] = (v8f)0.0f;

        #pragma unroll 4
        for (int ks = 0; ks < 144; ++ks) {
            int po = (2 * ks + half) * 16;                // per-lane pair offset
            v2f av0 = A0[po];
            v2f av1 = A0[po + 1 * PAIRS * 16];
            v2f av2 = A0[po + 2 * PAIRS * 16];
            v2f av3 = A0[po + 3 * PAIRS * 16];
            v2f bv0 = Bp0[po];
            v2f bv1 = Bp1[po];
            acc[0][0] = __builtin_amdgcn_wmma_f32_16x16x4_f32(false, av0, false, bv0, (short)0, acc[0][0], false, false);
            acc[0][1] = __builtin_amdgcn_wmma_f32_16x16x4_f32(false, av1, false, bv0, (short)0, acc[0][1], false, false);
            acc[0][2] = __builtin_amdgcn_wmma_f32_16x16x4_f32(false, av2, false, bv0, (short)0, acc[0][2], false, false);
            acc[0][3] = __builtin_amdgcn_wmma_f32_16x16x4_f32(false, av3, false, bv0, (short)0, acc[0][3], false, false);
            acc[1][0] = __builtin_amdgcn_wmma_f32_16x16x4_f32(false, av0, false, bv1, (short)0, acc[1][0], false, false);
            acc[1][1] = __builtin_amdgcn_wmma_f32_16x16x4_f32(false, av1, false, bv1, (short)0, acc[1][1], false, false);
            acc[1][2] = __builtin_amdgcn_wmma_f32_16x16x4_f32(false, av2, false, bv1, (short)0, acc[1][2], false, false);
            acc[1][3] = __builtin_amdgcn_wmma_f32_16x16x4_f32(false, av3, false, bv1, (short)0, acc[1][3], false, false);
        }
        // D layout: acc[n][t][r] = R[row = (lt0+t)*16 + r + 8*half][col = (mT0+n)*16 + n16]
        // rows scanned in increasing order; strict '>' keeps first max (jnp.argmax semantics)
        #pragma unroll
        for (int t = 0; t < MT; ++t) {
            #pragma unroll
            for (int r = 0; r < 8; ++r) {
                int l = (lt0 + t) * 16 + r + 8 * half;
                float v0 = acc[0][t][r];
                float v1 = acc[1][t][r];
                if (v0 > best0) { best0 = v0; bi0 = l; }
                if (v1 > best1) { best1 = v1; bi1 = l; }
            }
        }
    }
    // merge the two half-lanes (rows 0-7 vs 8-15 of each tile) — wave32 shuffle
    float ob0 = __shfl_xor(best0, 16, 32);
    int   oi0 = __shfl_xor(bi0, 16, 32);
    float ob1 = __shfl_xor(best1, 16, 32);
    int   oi1 = __shfl_xor(bi1, 16, 32);
    if (ob0 > best0) { best0 = ob0; bi0 = oi0; }          // other half holds larger rows
    if (ob1 > best1) { best1 = ob1; bi1 = oi1; }
    if (lane < 16) {
        size_t base = (size_t)seg * BL + b * L_;
        int m0 = mT0 * 16 + n16;
        int m1 = (mT0 + 1) * 16 + n16;
        pBest[base + m0] = best0;  pIdx[base + m0] = bi0;
        pBest[base + m1] = best1;  pIdx[base + m1] = bi1;
    }
}

__global__ void k_corr_reduce(const float* __restrict__ pBest, const int* __restrict__ pIdx,
                              const float* __restrict__ inv1,
                              float* __restrict__ Rstar, int* __restrict__ Rarg) {
    int i = blockIdx.x * 256 + threadIdx.x;               // over B*L
    float best = pBest[i];
    int idx = pIdx[i];
    #pragma unroll
    for (int s = 1; s < SEG; ++s) {                       // segments ordered by increasing l
        float bs = pBest[(size_t)s * BL + i];
        int   is = pIdx [(size_t)s * BL + i];
        if (bs > best) { best = bs; idx = is; }
    }
    Rstar[i] = best * inv1[i];
    Rarg[i] = idx;
}

// ---------------- gather best-match texture + fold(3,pad=1)/9 ----------------
__global__ void k_gather_fold(const float* __restrict__ f0hw, const int* __restrict__ Rarg,
                              float* __restrict__ T) {
    int i = blockIdx.x * 256 + threadIdx.x;               // over B*C*L
    int l = i & (L_ - 1);
    int bc = i >> 12;
    int b = bc >> 6, c = bc & 63;
    int y = l >> 6, x = l & 63;
    const float* img = f0hw + ((size_t)(b * C_ + c)) * L_;
    const int* ra = Rarg + b * L_;
    float s = 0.f;
    #pragma unroll
    for (int ki = 0; ki < 3; ++ki) {
        int my = y + 1 - ki; if ((unsigned)my >= 64u) continue;
        #pragma unroll
        for (int kj = 0; kj < 3; ++kj) {
            int mx = x + 1 - kj; if ((unsigned)mx >= 64u) continue;
            int a = ra[my * 64 + mx];
            int ay = a >> 6, ax = a & 63;
            int fy = ay + ki - 1, fx = ax + kj - 1;
            if ((unsigned)fy < 64u && (unsigned)fx < 64u) s += img[fy * 64 + fx];
        }
    }
    T[i] = s * (1.0f / 9.0f);
}

// ---------------- direct convs (small share of FLOPs; VALU) ----------------
__global__ void k_conv5(const float* __restrict__ inA, const float* __restrict__ inB,
                        const float* __restrict__ w, const float* __restrict__ bias,
                        const float* __restrict__ addp, float* __restrict__ out) {
    int b = blockIdx.z, co = blockIdx.y;
    int x = threadIdx.x & 63;
    int y = blockIdx.x * 4 + (threadIdx.x >> 6);
    int cin = inB ? 128 : 64;
    float s = bias[co];
    const float* wco = w + (size_t)co * cin * 25;
    for (int ci = 0; ci < cin; ++ci) {
        const float* img = (ci < 64) ? inA + ((size_t)(b * C_ + ci)) * L_
                                     : inB + ((size_t)(b * C_ + ci - 64)) * L_;
        const float* wc = wco + ci * 25;
        #pragma unroll
        for (int ky = 0; ky < 5; ++ky) {
            int iy = y + ky - 2; if ((unsigned)iy >= 64u) continue;
            const float* row = img + iy * 64;
            #pragma unroll
            for (int kx = 0; kx < 5; ++kx) {
                int ix = x + kx - 2;
                if ((unsigned)ix < 64u) s += row[ix] * wc[ky * 5 + kx];
            }
        }
    }
    size_t o = ((size_t)(b * C_ + co)) * L_ + y * 64 + x;
    if (addp) s += addp[o];
    out[o] = s;
}

__global__ void k_conv3(const float* __restrict__ in, const float* __restrict__ w,
                        const float* __restrict__ bias, const float* __restrict__ addp,
                        int relu, float* __restrict__ out) {
    int b = blockIdx.z, co = blockIdx.y;
    int x = threadIdx.x & 63;
    int y = blockIdx.x * 4 + (threadIdx.x >> 6);
    float s = bias[co];
    const float* wco = w + (size_t)co * C_ * 9;
    for (int ci = 0; ci < C_; ++ci) {
        const float* img = in + ((size_t)(b * C_ + ci)) * L_;
        const float* wc = wco + ci * 9;
        #pragma unroll
        for (int ky = 0; ky < 3; ++ky) {
            int iy = y + ky - 1; if ((unsigned)iy >= 64u) continue;
            const float* row = img + iy * 64;
            #pragma unroll
            for (int kx = 0; kx < 3; ++kx) {
                int ix = x + kx - 1;
                if ((unsigned)ix < 64u) s += row[ix] * wc[ky * 3 + kx];
            }
        }
    }
    size_t o = ((size_t)(b * C_ + co)) * L_ + y * 64 + x;
    if (addp) s += addp[o];
    if (relu) s = fmaxf(s, 0.f);
    out[o] = s;
}

// ---------------- Dfea + sq*S, then [B,C,L] -> [B,L,C] via LDS tile ----------------
__global__ void k_final(const float* __restrict__ Dfea, const float* __restrict__ sq,
                        const float* __restrict__ Rstar, float* __restrict__ out) {
    __shared__ float tile[64 * 65];
    int b = blockIdx.y, l0 = blockIdx.x * 64, t = threadIdx.x;
    #pragma unroll
    for (int rep = 0; rep < 16; ++rep) {
        int e = rep * 256 + t;
        int c = e >> 6, i = e & 63;                       // coalesced read over l
        size_t idx = ((size_t)(b * C_ + c)) * L_ + l0 + i;
        tile[c * 65 + i] = Dfea[idx] + sq[idx] * Rstar[b * L_ + l0 + i];
    }
    __syncthreads();
    #pragma unroll
    for (int rep = 0; rep < 16; ++rep) {
        int e = rep * 256 + t;
        int i = e >> 6, c = e & 63;                       // coalesced write over c
        out[((size_t)b * L_ + l0 + i) * C_ + c] = tile[c * 65 + i];
    }
}

extern "C" void kernel_launch(void* const* d_in, const int* in_sizes, int n_in,
                              void* d_out, int out_size, void* d_ws, size_t ws_size,
                              hipStream_t stream) {
    const float* c0     = (const float*)d_in[0];
    const float* f0     = (const float*)d_in[1];
    const float* c1     = (const float*)d_in[2];
    const float* head_w = (const float*)d_in[3];
    const float* head_b = (const float*)d_in[4];
    const float* rb_w1  = (const float*)d_in[5];
    const float* rb_b1  = (const float*)d_in[6];
    const float* rb_w2  = (const float*)d_in[7];
    const float* rb_b2  = (const float*)d_in[8];
    const float* tail_w = (const float*)d_in[9];
    const float* tail_b = (const float*)d_in[10];
    const float* sq_w   = (const float*)d_in[11];
    const float* sq_b   = (const float*)d_in[12];

    float* ws   = (float*)d_ws;
    float* c0hw = ws;                 ws += BCL;
    float* c1hw = ws;                 ws += BCL;
    float* f0hw = ws;                 ws += BCL;
    float* dC   = ws;                 ws += BCL;
    float* inv0 = ws;                 ws += BL;
    float* inv1 = ws;                 ws += BL;
    float* Rst  = ws;                 ws += BL;
    int*   Rarg = (int*)ws;           ws += BL;
    float* Tm   = ws;                 ws += BCL;
    float* U0n  = ws;                                   // 2 * B*L*576 floats (72 MB)
    float* U1   = U0n + (size_t)BL * K9;
    float* pBest = U1 + (size_t)BL * K9;
    int*   pIdx  = (int*)(pBest + (size_t)SEG * BL);
    // conv temporaries overlay the U region (only used after k_corr has drained, same stream)
    float* x1   = U0n;
    float* ht   = U0n + (size_t)BCL;
    float* xr   = U0n + 2 * (size_t)BCL;
    float* Dfea = U0n + 3 * (size_t)BCL;
    float* sqo  = U0n + 4 * (size_t)BCL;

    dim3 tb(256);
    dim3 gT(64, B_);
    k_transpose<<<gT, tb, 0, stream>>>(c0, c0hw);
    k_transpose<<<gT, tb, 0, stream>>>(c1, c1hw);
    k_transpose<<<gT, tb, 0, stream>>>(f0, f0hw);
    k_sub<<<(BCL + 255) / 256, tb, 0, stream>>>(c1hw, c0hw, dC, BCL);
    k_patchnorm<<<BL / 256, tb, 0, stream>>>(c0hw, inv0);
    k_patchnorm<<<BL / 256, tb, 0, stream>>>(c1hw, inv1);
    k_pack<<<dim3(LT_, B_), tb, 0, stream>>>(c0hw, inv0, U0n);    // row-normalized A
    k_pack<<<dim3(LT_, B_), tb, 0, stream>>>(c1hw, nullptr, U1);  // raw B
    k_corr<<<dim3(LT_ / (4 * NT), SEG, B_), dim3(128), 0, stream>>>(U0n, U1, pBest, pIdx);
    k_corr_reduce<<<BL / 256, tb, 0, stream>>>(pBest, pIdx, inv1, Rst, Rarg);
    k_gather_fold<<<BCL / 256, tb, 0, stream>>>(f0hw, Rarg, Tm);

    dim3 gC(16, C_, B_);
    // head
    k_conv5<<<gC, tb, 0, stream>>>(dC, nullptr, head_w, head_b, nullptr, x1);
    // only residual block 3 reaches the output (blocks 0-2 are dead in the reference)
    k_conv3<<<gC, tb, 0, stream>>>(x1, rb_w1 + 3 * C_ * C_ * 9, rb_b1 + 3 * C_, nullptr, 1, ht);
    k_conv3<<<gC, tb, 0, stream>>>(ht, rb_w2 + 3 * C_ * C_ * 9, rb_b2 + 3 * C_, x1, 0, xr);
    // tail, fused "+ x1" gives Delta_c_fea directly
    k_conv5<<<gC, tb, 0, stream>>>(xr, nullptr, tail_w, tail_b, x1, Dfea);
    // squeeze conv over concat(Dfea, T) without materializing the concat
    k_conv5<<<gC, tb, 0, stream>>>(Dfea, Tm, sq_w, sq_b, nullptr, sqo);
    // out = Dfea + sqo * S, transposed back to [B, L, C]
    k_final<<<gT, tb, 0, stream>>>(Dfea, sqo, Rst, (float*)d_out);
}